// RadioMambaNet_35150012351235
// MI455X (gfx1250) — compile-verified
//
#include <hip/hip_runtime.h>
#include <math.h>

// ---------------------------------------------------------------------------
// RadioMambaNet on MI455X (gfx1250): dense projections via fp32 WMMA
// (V_WMMA_F32_16X16X4_F32), dual-N-tile waves, b128 LDS staging, A prefetch.
// Selective scan and pointwise ops on VALU.
// ---------------------------------------------------------------------------

typedef float v2f __attribute__((ext_vector_type(2)));
typedef float v8f __attribute__((ext_vector_type(8)));

#define DSTATE 16

// ---------------- device helpers ----------------
__device__ __forceinline__ float siluf(float x) { return x / (1.f + expf(-x)); }
__device__ __forceinline__ float softplusf(float x) { return x > 20.f ? x : log1pf(expf(x)); }
__device__ __forceinline__ float gelu_exactf(float x) {
    return 0.5f * x * (1.f + erff(x * 0.70710678118654752f));
}

// ---------------------------------------------------------------------------
// WMMA GEMM: Y[M,N] = X[M,K] * W[N,K]^T (+bias, optional softplus)
//   - block = 128 threads = 4 waves; block supertile = 64 rows x 32 cols
//   - wave w -> rows [m0+16w, m0+16w+16); every wave covers BOTH 16-col tiles
//     with two accumulators, so one A fragment feeds two WMMAs
//   - REQUIRES M % 64 == 0 (true for every GEMM in this net) -> no M guards
//   - W chunk (32 x 64) staged in LDS via float4 (b128) loads/stores,
//     zero-padded at N/K edges -> WMMA inner loop is completely branch-free
//   - A fragments: aligned v2f global loads (lda always even here)
// A layout: lane r=lane%16 holds row m0+r; vgpr j holds K = 2*(lane/16)+j.
// B mirrors A with N; C/D: vgpr v, lane -> (m0 + v + 8*(lane/16), n0 + r).
// ---------------------------------------------------------------------------
#define GEMM_KC 64
#define GEMM_NT 32

__global__ __launch_bounds__(128) void gemm_wmma_lds(
    const float* __restrict__ X, int lda,
    const float* __restrict__ W, int ldw,
    const float* __restrict__ bias,
    float* __restrict__ Y, int ldc,
    int M, int N, int K, int act)
{
    __shared__ float Ws[GEMM_NT * GEMM_KC];   // 8 KB

    const int tid  = threadIdx.x;
    const int wave = tid >> 5;
    const int lane = tid & 31;
    const int r    = lane & 15;
    const int half = lane >> 4;

    const int n0 = blockIdx.x * GEMM_NT;
    const int m0 = blockIdx.y * 64 + wave * 16;

    v8f acc0 = {0.f, 0.f, 0.f, 0.f, 0.f, 0.f, 0.f, 0.f};
    v8f acc1 = {0.f, 0.f, 0.f, 0.f, 0.f, 0.f, 0.f, 0.f};

    const float* Xp = X + (size_t)(m0 + r) * lda;   // always in range (M % 64 == 0)
    const bool w4 = ((ldw & 3) == 0);               // float4-able weight rows

    for (int kc0 = 0; kc0 < K; kc0 += GEMM_KC) {
        __syncthreads();
        // ---- stage W[n0..n0+32) x [kc0..kc0+64) into LDS (b128 granularity) ----
        const bool full = (n0 + GEMM_NT <= N) && (kc0 + GEMM_KC <= K) && w4;
        if (full) {
#pragma unroll
            for (int f = tid; f < GEMM_NT * 16; f += 128) {
                const int row = f >> 4, c4 = (f & 15) << 2;
                *(float4*)&Ws[row * GEMM_KC + c4] =
                    *(const float4*)&W[(size_t)(n0 + row) * ldw + kc0 + c4];
            }
        } else {
            for (int f = tid; f < GEMM_NT * 16; f += 128) {
                const int row = f >> 4, c4 = (f & 15) << 2;
                const int gn = n0 + row;
                float4 v; v.x = v.y = v.z = v.w = 0.f;
                if (gn < N) {
                    const float* wp = W + (size_t)gn * ldw + kc0 + c4;
                    const int rem = K - (kc0 + c4);
                    if (rem >= 4 && w4) v = *(const float4*)wp;
                    else {
                        if (rem > 0) v.x = wp[0];
                        if (rem > 1) v.y = wp[1];
                        if (rem > 2) v.z = wp[2];
                        if (rem > 3) v.w = wp[3];
                    }
                }
                *(float4*)&Ws[row * GEMM_KC + c4] = v;
            }
        }
        __syncthreads();

        // prefetch next A chunk while computing on this one
        if (kc0 + GEMM_KC < K)
            __builtin_prefetch(Xp + kc0 + GEMM_KC, 0, 1);

        const int kend  = (K - kc0 < GEMM_KC) ? (K - kc0) : GEMM_KC;
        const int kmain = kend & ~3;   // trip counts are multiples of 4 steps here

        // ---- branch-free main loop: 1 A load feeds 2 WMMAs ----
#pragma unroll 4
        for (int kk = 0; kk < kmain; kk += 4) {
            const int ka = kk + half * 2;
            v2f a  = *(const v2f*)(Xp + kc0 + ka);
            v2f b0 = *(const v2f*)(&Ws[r * GEMM_KC + ka]);
            v2f b1 = *(const v2f*)(&Ws[(16 + r) * GEMM_KC + ka]);
            acc0 = __builtin_amdgcn_wmma_f32_16x16x4_f32(
                       false, a, false, b0, (short)0, acc0, false, false);
            acc1 = __builtin_amdgcn_wmma_f32_16x16x4_f32(
                       false, a, false, b1, (short)0, acc1, false, false);
        }
        // ---- rare K remainder (only dt_proj at dim 32: K == 2) ----
        if (kmain < kend) {
            const int ka = kmain + half * 2;
            v2f a, b0, b1;
            a.x = (ka     < kend) ? Xp[kc0 + ka]     : 0.f;
            a.y = (ka + 1 < kend) ? Xp[kc0 + ka + 1] : 0.f;
            b0 = *(const v2f*)(&Ws[r * GEMM_KC + ka]);          // LDS zero-padded
            b1 = *(const v2f*)(&Ws[(16 + r) * GEMM_KC + ka]);
            acc0 = __builtin_amdgcn_wmma_f32_16x16x4_f32(
                       false, a, false, b0, (short)0, acc0, false, false);
            acc1 = __builtin_amdgcn_wmma_f32_16x16x4_f32(
                       false, a, false, b1, (short)0, acc1, false, false);
        }
    }

    const int col0 = n0 + r;
    const int col1 = n0 + 16 + r;
    if (col0 < N) {
        const float bs = bias ? bias[col0] : 0.f;
#pragma unroll
        for (int v = 0; v < 8; ++v) {
            const int row = m0 + v + half * 8;   // always < M
            float val = acc0[v] + bs;
            if (act == 1) val = softplusf(val);
            Y[(size_t)row * ldc + col0] = val;
        }
    }
    if (col1 < N) {
        const float bs = bias ? bias[col1] : 0.f;
#pragma unroll
        for (int v = 0; v < 8; ++v) {
            const int row = m0 + v + half * 8;
            float val = acc1[v] + bs;
            if (act == 1) val = softplusf(val);
            Y[(size_t)row * ldc + col1] = val;
        }
    }
}

// ---------------- elementwise / small kernels ----------------
__global__ void fill_zero(float* __restrict__ p, int n) {
    int i = blockIdx.x * blockDim.x + threadIdx.x;
    if (i < n) p[i] = 0.f;
}

// per-pixel LayerNorm over channels (biased var), NHWC rows of width C
__global__ void ln_channel(const float* __restrict__ x, float* __restrict__ xn,
                           const float* __restrict__ g, const float* __restrict__ b,
                           int total, int C)
{
    int i = blockIdx.x * blockDim.x + threadIdx.x;
    if (i >= total) return;
    const float* row = x + (size_t)i * C;
    float mu = 0.f;
    for (int c = 0; c < C; ++c) mu += row[c];
    mu /= (float)C;
    float var = 0.f;
    for (int c = 0; c < C; ++c) { float d = row[c] - mu; var += d * d; }
    var /= (float)C;
    const float inv = rsqrtf(var + 1e-5f);
    float* out = xn + (size_t)i * C;
    for (int c = 0; c < C; ++c) out[c] = (row[c] - mu) * inv * g[c] + b[c];
}

// causal (dir=0) / anti-causal (dir=1) depthwise conv1d over L + bias + SiLU
// input = first di columns of xz (row stride 2*di)
__global__ void conv1d_silu(const float* __restrict__ xz, const float* __restrict__ w,
                            const float* __restrict__ bc, float* __restrict__ xc,
                            int Bn, int L, int di, int dir)
{
    int i = blockIdx.x * blockDim.x + threadIdx.x;
    int total = Bn * L * di;
    if (i >= total) return;
    int d = i % di;
    int l = (i / di) % L;
    int b = i / (di * L);
    float s = bc[d];
#pragma unroll
    for (int t = 0; t < 4; ++t) {
        int off = dir ? (3 - t) : (t - 3);
        int ll = l + off;
        if (ll >= 0 && ll < L)
            s += xz[((size_t)(b * L + ll)) * (2 * di) + d] * w[d * 4 + t];
    }
    xc[(size_t)(b * L + l) * di + d] = siluf(s);
}

// selective scan: one thread per (b, d); 16 states in registers.
// accumulates y (incl. xc*D) into yb for both directions.
__global__ void scan_seq(const float* __restrict__ xc, const float* __restrict__ dt,
                         const float* __restrict__ xd, int ldxd, int dr,
                         const float* __restrict__ A_log, const float* __restrict__ Dp,
                         float* __restrict__ yb, int Bn, int L, int di, int dir)
{
    int i = blockIdx.x * blockDim.x + threadIdx.x;
    if (i >= Bn * di) return;
    int b = i / di, d = i % di;
    float A[DSTATE], h[DSTATE];
#pragma unroll
    for (int s = 0; s < DSTATE; ++s) { A[s] = -expf(A_log[d * DSTATE + s]); h[s] = 0.f; }
    const float Dd = Dp[d];
    for (int t = 0; t < L; ++t) {
        const int l = dir ? (L - 1 - t) : t;
        const size_t p = (size_t)(b * L + l);
        const float dtv = dt[p * di + d];
        const float xv = xc[p * di + d];
        const float* row = xd + p * ldxd + dr;
        float y = Dd * xv;
#pragma unroll
        for (int s = 0; s < DSTATE; ++s) {
            const float a = expf(dtv * A[s]);
            h[s] = a * h[s] + (dtv * row[s]) * xv;
            y += h[s] * row[DSTATE + s];
        }
        yb[p * di + d] += y;
    }
}

// yb *= silu(z); z = cols [di, 2di) of xz
__global__ void gate_silu(float* __restrict__ yb, const float* __restrict__ xz,
                          int total, int di)
{
    int i = blockIdx.x * blockDim.x + threadIdx.x;
    if (i >= total * di) return;
    int d = i % di;
    int p = i / di;
    float z = xz[(size_t)p * (2 * di) + di + d];
    yb[i] *= siluf(z);
}

// depthwise 3x3 (pad 1) on NHWC + BN + exact GELU
__global__ void dw3x3_bn_gelu(const float* __restrict__ x, const float* __restrict__ w,
                              const float* __restrict__ g, const float* __restrict__ bb,
                              const float* __restrict__ mm, const float* __restrict__ vv,
                              float* __restrict__ y, int Bn, int H, int W, int C)
{
    int i = blockIdx.x * blockDim.x + threadIdx.x;
    int total = Bn * H * W * C;
    if (i >= total) return;
    int c = i % C;
    int ww = (i / C) % W;
    int hh = (i / (C * W)) % H;
    int b = i / (C * W * H);
    float s = 0.f;
#pragma unroll
    for (int ky = 0; ky < 3; ++ky)
#pragma unroll
        for (int kx = 0; kx < 3; ++kx) {
            int ih = hh + ky - 1, iw = ww + kx - 1;
            if (ih >= 0 && ih < H && iw >= 0 && iw < W)
                s += x[((size_t)((b * H + ih) * W + iw)) * C + c] * w[c * 9 + ky * 3 + kx];
        }
    float bn = (s - mm[c]) * rsqrtf(vv[c] + 1e-5f) * g[c] + bb[c];
    y[i] = gelu_exactf(bn);
}

// out = m + x + bn2(t2)   (elementwise, in place over x allowed)
__global__ void resid_add(const float* __restrict__ m, const float* __restrict__ x,
                          const float* __restrict__ t2,
                          const float* __restrict__ g, const float* __restrict__ bb,
                          const float* __restrict__ mm, const float* __restrict__ vv,
                          float* __restrict__ out, int total, int C)
{
    int i = blockIdx.x * blockDim.x + threadIdx.x;
    if (i >= total) return;
    int c = i % C;
    float bn = (t2[i] - mm[c]) * rsqrtf(vv[c] + 1e-5f) * g[c] + bb[c];
    out[i] = m[i] + x[i] + bn;
}

// patch embed: 3x3 conv pad 1, NCHW input -> NHWC output
__global__ void patch3x3(const float* __restrict__ x, const float* __restrict__ w,
                         const float* __restrict__ bias, float* __restrict__ y,
                         int Bn, int H, int W, int Ci, int Co)
{
    int i = blockIdx.x * blockDim.x + threadIdx.x;
    int total = Bn * H * W * Co;
    if (i >= total) return;
    int co = i % Co;
    int ww = (i / Co) % W;
    int hh = (i / (Co * W)) % H;
    int b = i / (Co * W * H);
    float s = bias[co];
    for (int ci = 0; ci < Ci; ++ci)
#pragma unroll
        for (int ky = 0; ky < 3; ++ky)
#pragma unroll
            for (int kx = 0; kx < 3; ++kx) {
                int ih = hh + ky - 1, iw = ww + kx - 1;
                if (ih >= 0 && ih < H && iw >= 0 && iw < W)
                    s += x[((size_t)(b * Ci + ci) * H + ih) * W + iw] *
                         w[((co * Ci + ci) * 3 + ky) * 3 + kx];
            }
    y[i] = s;
}

// 2x2 stride-2 downsample conv, NHWC
__global__ void down2x2(const float* __restrict__ x, const float* __restrict__ w,
                        const float* __restrict__ bias, float* __restrict__ y,
                        int Bn, int Hi, int Wi, int Ci, int Co)
{
    int Ho = Hi / 2, Wo = Wi / 2;
    int i = blockIdx.x * blockDim.x + threadIdx.x;
    int total = Bn * Ho * Wo * Co;
    if (i >= total) return;
    int co = i % Co;
    int wo = (i / Co) % Wo;
    int ho = (i / (Co * Wo)) % Ho;
    int b = i / (Co * Wo * Ho);
    float s = bias[co];
    for (int ci = 0; ci < Ci; ++ci)
#pragma unroll
        for (int ky = 0; ky < 2; ++ky)
#pragma unroll
            for (int kx = 0; kx < 2; ++kx)
                s += x[((size_t)((b * Hi + 2 * ho + ky) * Wi + 2 * wo + kx)) * Ci + ci] *
                     w[((co * Ci + ci) * 2 + ky) * 2 + kx];
    y[i] = s;
}

// 2x2 stride-2 transposed conv (each output pixel from exactly one input pixel), NHWC
__global__ void up2x2(const float* __restrict__ x, const float* __restrict__ w,
                      const float* __restrict__ bias, float* __restrict__ y,
                      int Bn, int Hi, int Wi, int Ci, int Co)
{
    int Ho = Hi * 2, Wo = Wi * 2;
    int i = blockIdx.x * blockDim.x + threadIdx.x;
    int total = Bn * Ho * Wo * Co;
    if (i >= total) return;
    int co = i % Co;
    int wo = (i / Co) % Wo;
    int ho = (i / (Co * Wo)) % Ho;
    int b = i / (Co * Wo * Ho);
    int hi = ho >> 1, wi = wo >> 1, ky = ho & 1, kx = wo & 1;
    float s = bias[co];
    const float* xp = x + ((size_t)((b * Hi + hi) * Wi + wi)) * Ci;
    for (int ci = 0; ci < Ci; ++ci)
        s += xp[ci] * w[((ci * Co + co) * 2 + ky) * 2 + kx];
    y[i] = s;
}

// copy src (total rows x C) into dst (total rows x dstC) at column colOff
__global__ void copy_cols(const float* __restrict__ src, float* __restrict__ dst,
                          int total, int C, int dstC, int colOff)
{
    int i = blockIdx.x * blockDim.x + threadIdx.x;
    if (i >= total * C) return;
    int c = i % C;
    int p = i / C;
    dst[(size_t)p * dstC + colOff + c] = src[i];
}

// final 1x1 conv to 1 channel, NHWC in -> NCHW out
__global__ void final1x1(const float* __restrict__ x, const float* __restrict__ w,
                         const float* __restrict__ bias, float* __restrict__ out,
                         int Bn, int H, int W, int C)
{
    int i = blockIdx.x * blockDim.x + threadIdx.x;
    int total = Bn * H * W;
    if (i >= total) return;
    const float* row = x + (size_t)i * C;
    float s = bias[0];
    for (int c = 0; c < C; ++c) s += row[c] * w[c];
    out[i] = s;
}

// ---------------------------------------------------------------------------
// host orchestration
// ---------------------------------------------------------------------------
struct BNP { const float *b, *g, *m, *v; };
struct MP  { const float *A_log, *D, *conv_b, *conv_w, *dt_b, *dt_w, *in_w,
                         *ln_b, *ln_w, *out_w, *xproj_w; };
struct BP  { BNP bn1, bn2; const float* dw_w; MP mamba; const float* pw_w; };
struct EncS { BP blocks[2]; const float *down_b, *down_w; };
struct DecS { BP blocks[2]; const float *fuse_b, *fuse_w, *up_b, *up_w; };

extern "C" void kernel_launch(void* const* d_in, const int* in_sizes, int n_in,
                              void* d_out, int out_size, void* d_ws, size_t ws_size,
                              hipStream_t stream)
{
    (void)in_sizes; (void)n_in; (void)out_size; (void)ws_size;
    const int Bn = 4;
    const int dims[4] = {32, 64, 128, 256};

    const float* x_in = (const float*)d_in[0];

    // ---- parse params (JAX pytree flatten: dict keys alphabetical) ----
    int idx = 1;
    auto nxt = [&]() { return (const float*)d_in[idx++]; };
    auto readBN = [&](BNP& p) { p.b = nxt(); p.g = nxt(); p.m = nxt(); p.v = nxt(); };
    auto readM = [&](MP& m) {
        m.A_log = nxt(); m.D = nxt(); m.conv_b = nxt(); m.conv_w = nxt();
        m.dt_b = nxt(); m.dt_w = nxt(); m.in_w = nxt(); m.ln_b = nxt();
        m.ln_w = nxt(); m.out_w = nxt(); m.xproj_w = nxt();
    };
    auto readBlk = [&](BP& b) {
        readBN(b.bn1); readBN(b.bn2); b.dw_w = nxt(); readM(b.mamba); b.pw_w = nxt();
    };

    BP bott[2];
    for (int i = 0; i < 2; ++i) readBlk(bott[i]);           // 'bott'
    DecS dec[3];
    for (int i = 0; i < 3; ++i) {                           // 'dec'
        readBlk(dec[i].blocks[0]); readBlk(dec[i].blocks[1]);
        dec[i].fuse_b = nxt(); dec[i].fuse_w = nxt();
        dec[i].up_b = nxt();   dec[i].up_w = nxt();
    }
    EncS enc[4];
    for (int i = 0; i < 4; ++i) {                           // 'enc'
        readBlk(enc[i].blocks[0]); readBlk(enc[i].blocks[1]);
        if (i < 3) { enc[i].down_b = nxt(); enc[i].down_w = nxt(); }
        else { enc[i].down_b = nullptr; enc[i].down_w = nullptr; }
    }
    const float* final_b = nxt(); const float* final_w = nxt();   // 'final'
    const float* patch_b = nxt(); const float* patch_w = nxt();   // 'patch'

    // ---- workspace bump allocator (floats) ----
    float* ws = (float*)d_ws;
    size_t cur = 0;
    auto alloc = [&](size_t n) { float* p = ws + cur; cur += (n + 63) & ~(size_t)63; return p; };
    float* ACT0 = alloc(524288);
    float* ACT1 = alloc(524288);
    float* SK0  = alloc(524288);
    float* SK1  = alloc(262144);
    float* SK2  = alloc(131072);
    float* XN   = alloc(524288);
    float* XZ   = alloc(2097152);
    float* XC   = alloc(1048576);
    float* XD   = alloc(786432);
    float* DT   = alloc(1048576);
    float* YB   = alloc(1048576);
    float* MB   = alloc(524288);
    float* T1   = alloc(524288);
    float* T2   = alloc(524288);
    float* CAT  = alloc(1048576);
    float* SKP[3] = {SK0, SK1, SK2};

    // M is a multiple of 64 for every GEMM in this network
    auto gemm = [&](const float* X, int lda, const float* W, int ldw, const float* bias,
                    float* Y, int ldc, int M, int N, int K, int act) {
        dim3 g((N + GEMM_NT - 1) / GEMM_NT, M / 64, 1);
        gemm_wmma_lds<<<g, dim3(128), 0, stream>>>(X, lda, W, ldw, bias, Y, ldc, M, N, K, act);
    };
    auto blk1d = [](int n) { return dim3((n + 255) / 256); };

    // ---- one mamba_conv_block, NHWC activation in `act`, in place ----
    auto runBlock = [&](float* act, int Hh, int C, const BP& p) {
        const int L = Hh * Hh, BL = Bn * L, di = 2 * C;
        const int dr = (C + 15) / 16, xd = dr + 2 * DSTATE;

        // --- mamba branch ---
        ln_channel<<<blk1d(BL), 256, 0, stream>>>(act, XN, p.mamba.ln_w, p.mamba.ln_b, BL, C);
        gemm(XN, C, p.mamba.in_w, C, nullptr, XZ, 2 * di, BL, 2 * di, C, 0);
        fill_zero<<<blk1d(BL * di), 256, 0, stream>>>(YB, BL * di);
        for (int dir = 0; dir < 2; ++dir) {
            conv1d_silu<<<blk1d(BL * di), 256, 0, stream>>>(XZ, p.mamba.conv_w, p.mamba.conv_b,
                                                            XC, Bn, L, di, dir);
            gemm(XC, di, p.mamba.xproj_w, di, nullptr, XD, xd, BL, xd, di, 0);
            gemm(XD, xd, p.mamba.dt_w, dr, p.mamba.dt_b, DT, di, BL, di, dr, 1 /*softplus*/);
            scan_seq<<<dim3((Bn * di + 63) / 64), 64, 0, stream>>>(XC, DT, XD, xd, dr,
                p.mamba.A_log, p.mamba.D, YB, Bn, L, di, dir);
        }
        gate_silu<<<blk1d(BL * di), 256, 0, stream>>>(YB, XZ, BL, di);
        gemm(YB, di, p.mamba.out_w, di, nullptr, MB, C, BL, C, di, 0);

        // --- conv branch + combine ---
        dw3x3_bn_gelu<<<blk1d(BL * C), 256, 0, stream>>>(act, p.dw_w,
            p.bn1.g, p.bn1.b, p.bn1.m, p.bn1.v, T1, Bn, Hh, Hh, C);
        gemm(T1, C, p.pw_w, C, nullptr, T2, C, BL, C, C, 0);
        resid_add<<<blk1d(BL * C), 256, 0, stream>>>(MB, act, T2,
            p.bn2.g, p.bn2.b, p.bn2.m, p.bn2.v, act, BL * C, C);
    };

    // ---- network ----
    patch3x3<<<blk1d(Bn * 64 * 64 * 32), 256, 0, stream>>>(x_in, patch_w, patch_b,
                                                           ACT0, Bn, 64, 64, 3, 32);
    float* act = ACT0;
    float* spare = ACT1;
    int Hh = 64;

    for (int s = 0; s < 4; ++s) {
        const int C = dims[s], L = Hh * Hh;
        if (s < 3)
            hipMemcpyAsync(SKP[s], act, (size_t)Bn * L * C * sizeof(float),
                           hipMemcpyDeviceToDevice, stream);
        runBlock(act, Hh, C, enc[s].blocks[0]);
        runBlock(act, Hh, C, enc[s].blocks[1]);
        if (s < 3) {
            const int Co = dims[s + 1], Ho = Hh / 2;
            down2x2<<<blk1d(Bn * Ho * Ho * Co), 256, 0, stream>>>(act, enc[s].down_w,
                enc[s].down_b, spare, Bn, Hh, Hh, C, Co);
            float* t = act; act = spare; spare = t;
            Hh = Ho;
        }
    }

    runBlock(act, Hh, 256, bott[0]);
    runBlock(act, Hh, 256, bott[1]);

    for (int i = 0; i < 3; ++i) {
        const int Ci = dims[3 - i], Co = Ci / 2, Ho = Hh * 2;
        up2x2<<<blk1d(Bn * Ho * Ho * Co), 256, 0, stream>>>(act, dec[i].up_w, dec[i].up_b,
                                                            spare, Bn, Hh, Hh, Ci, Co);
        copy_cols<<<blk1d(Bn * Ho * Ho * Co), 256, 0, stream>>>(spare, CAT,
            Bn * Ho * Ho, Co, 2 * Co, 0);
        copy_cols<<<blk1d(Bn * Ho * Ho * Co), 256, 0, stream>>>(SKP[2 - i], CAT,
            Bn * Ho * Ho, Co, 2 * Co, Co);
        gemm(CAT, 2 * Co, dec[i].fuse_w, 2 * Co, dec[i].fuse_b, act, Co,
             Bn * Ho * Ho, Co, 2 * Co, 0);
        Hh = Ho;
        runBlock(act, Hh, Co, dec[i].blocks[0]);
        runBlock(act, Hh, Co, dec[i].blocks[1]);
    }

    final1x1<<<blk1d(Bn * 64 * 64), 256, 0, stream>>>(act, final_w, final_b,
                                                      (float*)d_out, Bn, 64, 64, 32);
}